// WignerCombiningSingleUnrolled_55525337203349
// MI455X (gfx1250) — compile-verified
//
#include <hip/hip_runtime.h>
#include <hip/hip_bf16.h>

// out[n] (7x7) = U * H[n] * U^T
//   H[n][e][e'] = X1[n, i1(e,e')] * X2[n, i2(e,e')] * mult[e*T+e']   (dense TxT)
//   U[mu][e]    = (mu_e == mu)  -- indicator matrices, staged dense in LDS
// Computed per-sample (one sample per wave32) with V_WMMA_F32_16X16X4_F32.
// A operands for stage A are built inline: each H element is consumed by
// exactly one (lane, k-step) slot, so no H tile is ever materialized.

#define TP    64          // padded T (actual T = 54 for L1=L2=LAMBD=3)
#define WAVES 4           // samples (waves) per block, wave32

typedef __attribute__((ext_vector_type(2))) float v2f;
typedef __attribute__((ext_vector_type(8))) float v8f;

__global__ __launch_bounds__(32 * WAVES)
void wigner_wmma_kernel(const float* __restrict__ X1g,
                        const float* __restrict__ X2g,
                        const float* __restrict__ mult,
                        const int* __restrict__ m1,  const int* __restrict__ m1p,
                        const int* __restrict__ m2,  const int* __restrict__ m2p,
                        const int* __restrict__ mub,
                        float* __restrict__ out, int N, int T)
{
    // block-shared, built once behind one barrier
    __shared__ __align__(16) float multS[TP][TP + 4]; // stride 68: conflict-free halves
    __shared__ __align__(16) float Ut[TP][16];        // U^T[e', mup] indicator
    __shared__ __align__(16) float Ua[TP][16];        // U[mu, e] indicator (transposed store)
    __shared__ int arow12[TP];                        // m1*7 | (m2*7)<<16   (row parts)
    __shared__ int bcol12[TP];                        // m1p  | m2p<<16      (col parts)
    // per-wave scratch
    __shared__ __align__(16) float Rs[WAVES][TP][18]; // stride 18: conflict-free halves
    __shared__ float X1s[WAVES][56];
    __shared__ float X2s[WAVES][56];

    const int tid  = threadIdx.x;
    const int w    = tid >> 5;       // wave id = sample slot
    const int lane = tid & 31;
    const int half = lane >> 4;      // 0: lanes 0-15, 1: lanes 16-31
    const int l16  = lane & 15;

    // ---- stage zero-padded mult (kills all guards in the hot loop) ----
    for (int idx = tid; idx < TP * TP; idx += blockDim.x) {
        const int e = idx >> 6, col = idx & (TP - 1);
        multS[e][col] = (e < T && col < T) ? mult[e * T + col] : 0.0f;
    }
    // ---- dense indicator matrices ----
    for (int idx = tid; idx < TP * 16; idx += blockDim.x) {
        const int e = idx >> 4, q = idx & 15;
        Ut[e][q] = (e < T && (mub[e] % 7) == q)     ? 1.0f : 0.0f;
        Ua[e][q] = (e < T && (mub[e * T] / 7) == q) ? 1.0f : 0.0f;
    }
    // ---- factored gather-index tables ----
    for (int e = tid; e < TP; e += blockDim.x) {
        if (e < T) {
            arow12[e] = (m1[e * T] * 7) | ((m2[e * T] * 7) << 16);
            bcol12[e] = m1p[e] | (m2p[e] << 16);
        } else {
            arow12[e] = 0; bcol12[e] = 0;     // safe indices; multS row is 0 anyway
        }
    }
    __syncthreads();

    const int n = blockIdx.x * WAVES + w;
    if (n >= N) return;                  // wave-uniform; no barriers after this

    // ---- stage sample rows of X1/X2 into LDS for dynamic gathers ----
    for (int j = lane; j < 49; j += 32) {
        X1s[w][j] = X1g[n * 49 + j];
        X2s[w][j] = X2g[n * 49 + j];
    }

    // ---- stage A: R = H * U^T, A operand computed inline per lane ----
    for (int rt = 0; rt < TP / 16; ++rt) {
        const int arow = rt * 16 + l16;          // my H row (A-matrix M index)
        const int ar   = arow12[arow];
        const int i1r  = ar & 0xffff;
        const int i2r  = ar >> 16;
        v8f acc = {};
        #pragma unroll 4
        for (int kk = 0; kk < TP; kk += 4) {
            const int kb  = kk + 2 * half;       // my K pair: kb, kb+1
            const int bc0 = bcol12[kb];
            const int bc1 = bcol12[kb + 1];
            v2f a, b;
            a.x = X1s[w][i1r + (bc0 & 0xffff)] * X2s[w][i2r + (bc0 >> 16)]
                  * multS[arow][kb];
            a.y = X1s[w][i1r + (bc1 & 0xffff)] * X2s[w][i2r + (bc1 >> 16)]
                  * multS[arow][kb + 1];
            b.x = Ut[kb][l16];
            b.y = Ut[kb + 1][l16];
            acc = __builtin_amdgcn_wmma_f32_16x16x4_f32(
                      false, a, false, b, (short)0, acc, false, false);
        }
        // C layout: VGPR r -> row (r + 8*half), col l16
        #pragma unroll
        for (int r = 0; r < 8; ++r)
            Rs[w][rt * 16 + r + 8 * half][l16] = acc[r];
    }

    // ---- stage B: out = U * R ----
    v8f acc2 = {};
    #pragma unroll 4
    for (int kk = 0; kk < TP; kk += 4) {
        const int kb = kk + 2 * half;
        v2f a, b;
        a.x = Ua[kb][l16];
        a.y = Ua[kb + 1][l16];
        b.x = Rs[w][kb][l16];
        b.y = Rs[w][kb + 1][l16];
        acc2 = __builtin_amdgcn_wmma_f32_16x16x4_f32(
                   false, a, false, b, (short)0, acc2, false, false);
    }

    // store valid 7x7 block: VGPR r -> mu = r + 8*half (only half==0, r<7), mup = l16
    if (half == 0 && l16 < 7) {
        #pragma unroll
        for (int r = 0; r < 7; ++r)
            out[n * 49 + r * 7 + l16] = acc2[r];
    }
}

// Scalar fallback (segment-accumulated; q-sequence is lane-uniform) in case
// n_aligned is not a perfect square <= TP^2.
__global__ void wigner_fallback_kernel(const float* __restrict__ X1g,
                                       const float* __restrict__ X2g,
                                       const float* __restrict__ mult,
                                       const int* __restrict__ m1,  const int* __restrict__ m1p,
                                       const int* __restrict__ m2,  const int* __restrict__ m2p,
                                       const int* __restrict__ mub,
                                       float* __restrict__ out, int N, int K)
{
    const int n = blockIdx.x * blockDim.x + threadIdx.x;
    if (n >= N) return;
    const float* x1 = X1g + n * 49;
    const float* x2 = X2g + n * 49;
    float* o = out + n * 49;
    for (int q = 0; q < 49; ++q) o[q] = 0.0f;
    int qcur = mub[0];
    float acc = 0.0f;
    for (int k = 0; k < K; ++k) {
        const int q = mub[k];
        if (q != qcur) { o[qcur] += acc; acc = 0.0f; qcur = q; }
        acc += x1[m1[k] * 7 + m1p[k]] * x2[m2[k] * 7 + m2p[k]] * mult[k];
    }
    o[qcur] += acc;
}

extern "C" void kernel_launch(void* const* d_in, const int* in_sizes, int n_in,
                              void* d_out, int out_size, void* d_ws, size_t ws_size,
                              hipStream_t stream) {
    const float* X1   = (const float*)d_in[0];
    const float* X2   = (const float*)d_in[1];
    const float* mult = (const float*)d_in[2];
    const int*   m1   = (const int*)d_in[3];
    const int*   m1p  = (const int*)d_in[4];
    const int*   m2   = (const int*)d_in[5];
    const int*   m2p  = (const int*)d_in[6];
    const int*   mub  = (const int*)d_in[7];
    float*       out  = (float*)d_out;

    const int N = in_sizes[0] / 49;       // 20000
    const int K = in_sizes[2];            // n_aligned (expect 54^2 = 2916)
    int T = 0;
    while ((T + 1) * (T + 1) <= K) ++T;   // integer sqrt

    if (T * T == K && T <= TP) {
        dim3 block(32 * WAVES);
        dim3 grid((N + WAVES - 1) / WAVES);
        wigner_wmma_kernel<<<grid, block, 0, stream>>>(
            X1, X2, mult, m1, m1p, m2, m2p, mub, out, N, T);
    } else {
        const int bs = 256;
        wigner_fallback_kernel<<<(N + bs - 1) / bs, bs, 0, stream>>>(
            X1, X2, mult, m1, m1p, m2, m2p, mub, out, N, K);
    }
}